// GAT_11484742549904
// MI455X (gfx1250) — compile-verified
//
#include <hip/hip_runtime.h>
#include <math.h>

// ---------------------------------------------------------------------------
// 3-layer GAT for MI455X (gfx1250).
// GEMMs use V_WMMA_F32_16X16X4_F32 (exact fp32 match to the reference), with
// the 16-row A strip staged in LDS (conflict-free padded layout) and shared
// by all 16 waves of the block -> kills the 16x redundant A traffic.
// Edge softmax/aggregation via L2 atomics (feat tensors are 51MB and fit in
// the 192MB L2; the phase is atomic/L2 bound, HBM at 23.3TB/s is idle).
// ---------------------------------------------------------------------------

typedef __attribute__((ext_vector_type(2))) float v2f;
typedef __attribute__((ext_vector_type(8))) float v8f;

#define HIDC 256
#define HEADS 4
#define DHC 64

__device__ __forceinline__ float atomicMaxF(float* addr, float value) {
  // monotonic int/uint mapping: works for mixed-sign floats incl. -inf init
  if (value >= 0.0f)
    return __int_as_float(atomicMax((int*)addr, __float_as_int(value)));
  else
    return __uint_as_float(atomicMin((unsigned int*)addr, __float_as_uint(value)));
}

__device__ __forceinline__ float lrelu(float v) {
  return v > 0.0f ? v : 0.2f * v;
}

// C[M x 256] = A[M x K] @ W[K x 256] (+ optional bias broadcast over rows).
// grid.x = ceil(M/16) row strips; block = 512 threads = 16 waves; wave w owns
// the 16-col tile at w*16. A strip lives in LDS, padded stride K+4 floats:
// 16 rows * 4-bank stride + kHalf offsets cover all 64 banks conflict-free
// for the per-lane ds_load_b64 fragment reads.
template <int K>
__global__ __launch_bounds__(512)
void gemm_wmma_f32(const float* __restrict__ A, const float* __restrict__ W,
                   const float* __restrict__ bias, float* __restrict__ C,
                   int M) {
  constexpr int LDA = K + 4;
  __shared__ float As[16 * LDA];

  const int tid   = threadIdx.x;
  const int mBase = blockIdx.x << 4;

  // cooperative stage: 16 x K floats, 512 threads -> K/32 elems each
  #pragma unroll
  for (int i = tid; i < 16 * K; i += 512) {
    int r = i / K;                      // K is a power of two -> shifts
    int c = i - r * K;
    int row = mBase + r;
    if (row >= M) row = M - 1;          // clamp (stores guarded below)
    As[r * LDA + c] = A[(long)row * K + c];
  }
  __syncthreads();

  const int lane  = tid & 31;
  const int wave  = tid >> 5;           // 0..15
  const int nBase = wave << 4;
  const int col   = lane & 15;
  const int kHalf = (lane >> 4) << 1;   // lanes 0-15 -> K 0/1, 16-31 -> K 2/3

  const float* as = &As[(lane & 15) * LDA + kHalf];
  const float* bp = W + (long)kHalf * HIDC + nBase + col;

  v8f acc = {};
  #pragma unroll 8
  for (int kb = 0; kb < K; kb += 4) {
    v2f a, b;
    a.x = as[0];            // A[row, kb+kHalf]     -> VGPR0 (ds_load_b64)
    a.y = as[1];            // A[row, kb+kHalf+1]   -> VGPR1
    b.x = bp[0];            // W[kb+kHalf,   nBase+col]
    b.y = bp[HIDC];         // W[kb+kHalf+1, nBase+col]
    acc = __builtin_amdgcn_wmma_f32_16x16x4_f32(
        /*neg_a=*/false, a, /*neg_b=*/false, b,
        /*c_mod=*/(short)0, acc, /*reuse_a=*/false, /*reuse_b=*/false);
    as += 4;
    bp += 4 * HIDC;
  }

  // C layout: vgpr r -> row mBase + r (+8 for lanes 16-31), col = nBase+lane&15
  const int rowOff = (lane >> 4) << 3;  // 0 or 8
  const float bc = bias ? bias[nBase + col] : 0.0f;
  #pragma unroll
  for (int r = 0; r < 8; ++r) {
    int row = mBase + rowOff + r;
    if (row < M) C[(long)row * HIDC + nBase + col] = acc[r] + bc;
  }
}

// Per (node,head) attention dots: el = <feat[n,h,:], al[h,:]>, er likewise.
// One wave32 per (n,h); coalesced 64-float segment, shuffle reduction.
__global__ __launch_bounds__(256)
void el_er_kernel(const float* __restrict__ feat, const float* __restrict__ al,
                  const float* __restrict__ ar, float* __restrict__ el,
                  float* __restrict__ er, int NH) {
  int warp = blockIdx.x * (blockDim.x >> 5) + (threadIdx.x >> 5);
  int lane = threadIdx.x & 31;
  if (warp >= NH) return;
  int n = warp >> 2, h = warp & 3;
  const float* f  = feat + (long)n * HIDC + h * DHC;
  const float* a1 = al + h * DHC;
  const float* a2 = ar + h * DHC;
  float f0 = f[lane], f1 = f[lane + 32];
  float sl = f0 * a1[lane] + f1 * a1[lane + 32];
  float sr = f0 * a2[lane] + f1 * a2[lane + 32];
  #pragma unroll
  for (int off = 16; off; off >>= 1) {
    sl += __shfl_down(sl, off, 32);
    sr += __shfl_down(sr, off, 32);
  }
  if (lane == 0) { el[warp] = sl; er[warp] = sr; }
}

__global__ void init_nh_kernel(float* __restrict__ m, float* __restrict__ dn,
                               int NH) {
  int i = blockIdx.x * blockDim.x + threadIdx.x;
  if (i < NH) { m[i] = -INFINITY; dn[i] = 0.0f; }
}

__global__ void init_out_kernel(const float* __restrict__ hin,
                                const float* __restrict__ bias,
                                float* __restrict__ out, long total) {
  long i = (long)blockIdx.x * blockDim.x + threadIdx.x;
  if (i < total) out[i] = hin[i] + bias[i & (HIDC - 1)];
}

__global__ void edge_max_kernel(const int* __restrict__ src,
                                const int* __restrict__ dst,
                                const float* __restrict__ el,
                                const float* __restrict__ er,
                                float* __restrict__ m, int EH) {
  int idx = blockIdx.x * blockDim.x + threadIdx.x;
  if (idx >= EH) return;
  int e = idx >> 2, h = idx & 3;
  int s = src[e], d = dst[e];
  float v = lrelu(el[s * 4 + h] + er[d * 4 + h]);
  atomicMaxF(&m[d * 4 + h], v);
}

__global__ void edge_exp_kernel(const int* __restrict__ src,
                                const int* __restrict__ dst,
                                const float* __restrict__ el,
                                const float* __restrict__ er,
                                const float* __restrict__ m,
                                float* __restrict__ dn,
                                float* __restrict__ ex, int EH) {
  int idx = blockIdx.x * blockDim.x + threadIdx.x;
  if (idx >= EH) return;
  int e = idx >> 2, h = idx & 3;
  int s = src[e], d = dst[e];
  float v = lrelu(el[s * 4 + h] + er[d * 4 + h]);
  float mm = m[d * 4 + h];
  if (!(mm > -1e38f)) mm = 0.0f;          // zero-in-degree safety (-inf -> 0)
  float x = __expf(v - mm);
  ex[idx] = x;
  atomicAdd(&dn[d * 4 + h], x);
}

// One 256-thread block per edge: thread c = h*64+d. Coalesced gather of
// feat[src], global_atomic_add_f32 scatter into out[dst] (pre-init res+bias).
__global__ __launch_bounds__(256)
void aggregate_kernel(const int* __restrict__ src, const int* __restrict__ dst,
                      const float* __restrict__ feat,
                      const float* __restrict__ ex,
                      const float* __restrict__ dn, float* __restrict__ out) {
  int e = blockIdx.x;
  int c = threadIdx.x;                    // 0..255
  int h = c >> 6;
  int s = src[e], d = dst[e];
  float denom = dn[d * 4 + h];
  float alpha = ex[e * 4 + h] / fmaxf(denom, 1e-16f);
  atomicAdd(&out[(long)d * HIDC + c], alpha * feat[(long)s * HIDC + c]);
}

__global__ void relu_kernel(const float* __restrict__ in, float* __restrict__ out,
                            long total) {
  long i = (long)blockIdx.x * blockDim.x + threadIdx.x;
  if (i < total) out[i] = fmaxf(in[i], 0.0f);
}

__global__ void mean_heads_kernel(const float* __restrict__ hbuf,
                                  float* __restrict__ out, long total) {
  long i = (long)blockIdx.x * blockDim.x + threadIdx.x;
  if (i >= total) return;
  long n = i >> 6;
  int d = (int)(i & 63);
  const float* p = hbuf + n * HIDC + d;
  out[i] = 0.25f * (p[0] + p[DHC] + p[2 * DHC] + p[3 * DHC]);
}

extern "C" void kernel_launch(void* const* d_in, const int* in_sizes, int n_in,
                              void* d_out, int out_size, void* d_ws, size_t ws_size,
                              hipStream_t stream) {
  const float* x     = (const float*)d_in[0];
  const int*   ei    = (const int*)d_in[1];
  const float* W[3]  = {(const float*)d_in[2], (const float*)d_in[3],
                        (const float*)d_in[4]};
  const float* al[3] = {(const float*)d_in[5], (const float*)d_in[6],
                        (const float*)d_in[7]};
  const float* ar[3] = {(const float*)d_in[8], (const float*)d_in[9],
                        (const float*)d_in[10]};
  const float* b[3]  = {(const float*)d_in[11], (const float*)d_in[12],
                        (const float*)d_in[13]};
  const float* resW0 = (const float*)d_in[14];

  const int N = in_sizes[0] / 128;        // nodes (F_IN = 128)
  const int E = in_sizes[1] / 2;          // edges
  const int* src = ei;
  const int* dst = ei + E;

  // workspace layout (floats)
  float* ws   = (float*)d_ws;
  float* FEAT = ws;                               // N*256
  float* OUT  = FEAT + (size_t)N * HIDC;          // N*256
  float* HBUF = OUT  + (size_t)N * HIDC;          // N*256
  float* ELP  = HBUF + (size_t)N * HIDC;          // N*4
  float* ERP  = ELP  + (size_t)N * HEADS;         // N*4
  float* MP   = ERP  + (size_t)N * HEADS;         // N*4
  float* DNP  = MP   + (size_t)N * HEADS;         // N*4
  float* EXP  = DNP  + (size_t)N * HEADS;         // E*4

  const int  mtiles = (N + 15) / 16;
  const long totNF  = (long)N * HIDC;
  const int  NH     = N * HEADS;
  const int  EH     = E * HEADS;
  const int  gNF    = (int)((totNF + 255) / 256);
  const int  gNH    = (NH + 255) / 256;
  const int  gEH    = (EH + 255) / 256;
  const int  gELR   = (NH + 7) / 8;               // 8 waves / block

  auto attention = [&](const float* alw, const float* arw) {
    el_er_kernel<<<gELR, 256, 0, stream>>>(FEAT, alw, arw, ELP, ERP, NH);
    init_nh_kernel<<<gNH, 256, 0, stream>>>(MP, DNP, NH);
    edge_max_kernel<<<gEH, 256, 0, stream>>>(src, dst, ELP, ERP, MP, EH);
    edge_exp_kernel<<<gEH, 256, 0, stream>>>(src, dst, ELP, ERP, MP, DNP, EXP, EH);
    aggregate_kernel<<<E, 256, 0, stream>>>(src, dst, FEAT, EXP, DNP, OUT);
  };

  // ---- layer 0: residual = x @ resW0 (+b0 fused into OUT init) ----
  gemm_wmma_f32<128><<<mtiles, 512, 0, stream>>>(x, W[0], nullptr, FEAT, N);
  gemm_wmma_f32<128><<<mtiles, 512, 0, stream>>>(x, resW0, b[0], OUT, N);
  attention(al[0], ar[0]);
  relu_kernel<<<gNF, 256, 0, stream>>>(OUT, HBUF, totNF);

  // ---- layer 1: identity residual ----
  gemm_wmma_f32<256><<<mtiles, 512, 0, stream>>>(HBUF, W[1], nullptr, FEAT, N);
  init_out_kernel<<<gNF, 256, 0, stream>>>(HBUF, b[1], OUT, totNF);
  attention(al[1], ar[1]);
  relu_kernel<<<gNF, 256, 0, stream>>>(OUT, HBUF, totNF);

  // ---- layer 2: identity residual, then mean over heads ----
  gemm_wmma_f32<256><<<mtiles, 512, 0, stream>>>(HBUF, W[2], nullptr, FEAT, N);
  init_out_kernel<<<gNF, 256, 0, stream>>>(HBUF, b[2], OUT, totNF);
  attention(al[2], ar[2]);
  mean_heads_kernel<<<(int)(((long)N * DHC + 255) / 256), 256, 0, stream>>>(
      OUT, (float*)d_out, (long)N * DHC);
}